// trunc_simple_73985106641583
// MI455X (gfx1250) — compile-verified
//
#include <hip/hip_runtime.h>
#include <stdint.h>

typedef __attribute__((ext_vector_type(4))) unsigned int u32x4;
typedef __attribute__((ext_vector_type(8))) unsigned int u32x8;
typedef __attribute__((ext_vector_type(4))) float        f32x4;

#define ROW_F 8192
#define NTHR  256

struct __align__(16) SmemT {
    unsigned int keys[ROW_F];   // 32 KB TDM landing zone (read once by transform)
    unsigned int histA[256];    // hi-select histogram (also shared level-0 histogram)
    unsigned int histB[256];    // lo-select histogram
    unsigned int bc[4];         // {prefHi, remHi, prefLo, remLo}
};

__device__ __forceinline__ unsigned sortKey(float f) {
    unsigned u = __float_as_uint(f);
    return u ^ ((unsigned)((int)u >> 31) | 0x80000000u);
}
__device__ __forceinline__ float unsortKey(unsigned k) {
    unsigned u = k ^ (((int)k >> 31) ? 0x80000000u : 0xFFFFFFFFu);
    return __uint_as_float(u);
}
__device__ __forceinline__ float maskVal(unsigned key, unsigned kHi, unsigned kLo) {
    float v = unsortKey(key);
    // zero the top-K and bottom-K (v*0 keeps reference's signed-zero behavior)
    return (key >= kHi || key <= kLo) ? v * 0.0f : v;
}
__device__ __forceinline__ unsigned laneId() {
    return __builtin_amdgcn_mbcnt_lo(~0u, 0u);   // wave32: lane index
}

// Wave-aggregated histogram increment: one atomicAdd per distinct bin per wave.
// Critical for N(0,1) data, whose key exponent byte collapses into ~3 bins.
__device__ __forceinline__ void waveAggInc(unsigned* hist, unsigned bin, bool pred) {
    while (__any(pred)) {                       // __any is wave-uniform -> uniform loop
        unsigned long long act = __ballot(pred);
        unsigned first = (unsigned)__ffsll((unsigned long long)act) - 1u;
        unsigned fbin  = __shfl(bin, (int)first, 32);
        bool same = pred && (bin == fbin);
        unsigned long long grp = __ballot(same);
        if (same) {
            if (laneId() == first)
                atomicAdd(&hist[fbin], (unsigned)__popcll(grp));
            pred = false;
        }
    }
}

__global__ void __launch_bounds__(NTHR)
topk_mask_rowselect_kernel(const float* __restrict__ x,
                           const float* __restrict__ w,
                           const int*   __restrict__ kp,
                           float*       __restrict__ out)
{
    __shared__ SmemT sm;
    const unsigned t   = threadIdx.x;
    const unsigned row = blockIdx.x;
    const int K = *kp;

    // ---- Stage the whole 32KB row into LDS with one Tensor Data Mover op ----
    if (t < 32) {   // wave 0 only (s_cbranch_execz skips waves 1..7)
        uint64_t ga = (uint64_t)(uintptr_t)(x + (size_t)row * ROW_F);
        u32x4 g0;
        g0[0] = 1u;                                           // count=1, user desc
        g0[1] = 0u;                                           // lds_addr = 0 (sm.keys)
        g0[2] = (unsigned)ga;                                 // global_addr[31:0]
        g0[3] = ((unsigned)(ga >> 32) & 0x01FFFFFFu)          // global_addr[56:32]
              | 0x80000000u;                                  // type = 2 ("image")
        u32x8 g1;
        g1[0] = 2u << 16;                                     // data_size = 4 bytes
        g1[1] = (ROW_F & 0xFFFFu) << 16;                      // tensor_dim0 lo16
        g1[2] = 1u << 16;                                     // tensor_dim0 hi=0 | tensor_dim1=1
        g1[3] = (ROW_F & 0xFFFFu) << 16;                      // tensor_dim1 hi=0 | tile_dim0=8192
        g1[4] = 1u;                                           // tile_dim1=1, tile_dim2=0
        g1[5] = ROW_F;                                        // tensor_dim0_stride lo32
        g1[6] = 0u;
        g1[7] = 0u;
        asm volatile("tensor_load_to_lds %0, %1" :: "s"(g0), "s"(g1) : "memory");
        __builtin_amdgcn_s_wait_tensorcnt(0);
    }
    if (t == 0) {
        sm.bc[0] = 0u; sm.bc[1] = (unsigned)K;   // prefHi, remHi
        sm.bc[2] = 0u; sm.bc[3] = (unsigned)K;   // prefLo, remLo
    }
    __syncthreads();   // TDM data + bc init visible to all waves

    // ---- Transform: raw x -> sortable key of (x*weight); keep keys in VGPRs ----
    u32x4 kreg[8];
    #pragma unroll
    for (int j = 0; j < 8; ++j) {
        unsigned vidx = (unsigned)j * NTHR + t;       // float4 index within row
        f32x4 wv = *(const f32x4*)(w + vidx * 4u);     // L2-resident after first rows
        u32x4 uk = *(const u32x4*)&sm.keys[vidx * 4u];
        uk[0] = sortKey(__uint_as_float(uk[0]) * wv[0]);
        uk[1] = sortKey(__uint_as_float(uk[1]) * wv[1]);
        uk[2] = sortKey(__uint_as_float(uk[2]) * wv[2]);
        uk[3] = sortKey(__uint_as_float(uk[3]) * wv[3]);
        kreg[j] = uk;
    }

    // ---- 4-level radix select: exact K-th largest & K-th smallest keys ----
    unsigned prefHi = 0u, prefLo = 0u;
    for (int lv = 0; lv < 4; ++lv) {
        sm.histA[t] = 0u;
        sm.histB[t] = 0u;
        __syncthreads();
        const int shift = 24 - 8 * lv;
        #pragma unroll
        for (int j = 0; j < 8; ++j) {
            u32x4 uk = kreg[j];
            #pragma unroll
            for (int e = 0; e < 4; ++e) {
                unsigned key = uk[e];
                if (lv == 0) {
                    // level 0: one histogram serves both selections
                    waveAggInc(sm.histA, key >> 24, true);
                } else {
                    const int sh8 = shift + 8;
                    unsigned bin = (key >> shift) & 255u;
                    waveAggInc(sm.histA, bin, ((key ^ prefHi) >> sh8) == 0u);
                    waveAggInc(sm.histB, bin, ((key ^ prefLo) >> sh8) == 0u);
                }
            }
        }
        __syncthreads();
        if (t == 0) {            // descending scan -> K-th largest (wave 0)
            unsigned rem = sm.bc[1], cum = 0u, bin = 0u;
            for (int b = 255; b >= 0; --b) {
                unsigned c = sm.histA[b];
                if (cum + c >= rem) { bin = (unsigned)b; break; }
                cum += c;
            }
            sm.bc[0] |= bin << shift;
            sm.bc[1] = rem - cum;
        }
        if (t == 64) {           // ascending scan -> K-th smallest (wave 2, concurrent)
            unsigned rem = sm.bc[3], cum = 0u, bin = 0u;
            const unsigned* h = (lv == 0) ? sm.histA : sm.histB;
            for (int b = 0; b < 256; ++b) {
                unsigned c = h[b];
                if (cum + c >= rem) { bin = (unsigned)b; break; }
                cum += c;
            }
            sm.bc[2] |= bin << shift;
            sm.bc[3] = rem - cum;
        }
        __syncthreads();
        prefHi = sm.bc[0];
        prefLo = sm.bc[2];
    }
    // prefHi/prefLo are now the exact keys of the K-th largest / K-th smallest.

    // ---- Final sweep from registers: zero top/bottom K, stream out (NT) ----
    f32x4* out4 = (f32x4*)(out + (size_t)row * ROW_F);
    #pragma unroll
    for (int j = 0; j < 8; ++j) {
        unsigned vidx = (unsigned)j * NTHR + t;
        u32x4 uk = kreg[j];
        f32x4 r;
        r[0] = maskVal(uk[0], prefHi, prefLo);
        r[1] = maskVal(uk[1], prefHi, prefLo);
        r[2] = maskVal(uk[2], prefHi, prefLo);
        r[3] = maskVal(uk[3], prefHi, prefLo);
        __builtin_nontemporal_store(r, out4 + vidx);   // output never re-read
    }
}

extern "C" void kernel_launch(void* const* d_in, const int* in_sizes, int n_in,
                              void* d_out, int out_size, void* d_ws, size_t ws_size,
                              hipStream_t stream) {
    (void)n_in; (void)d_ws; (void)ws_size; (void)out_size;
    const float* x  = (const float*)d_in[0];
    const float* w  = (const float*)d_in[1];
    const int*   kp = (const int*)d_in[2];
    float*       out = (float*)d_out;
    const int F = in_sizes[1];            // 8192 (kernel specialized to this width)
    const int B = in_sizes[0] / F;        // 8192 rows
    topk_mask_rowselect_kernel<<<dim3(B), dim3(NTHR), 0, stream>>>(x, w, kp, out);
}